// PositionalEncoding_47648367182212
// MI455X (gfx1250) — compile-verified
//
#include <hip/hip_runtime.h>
#include <math.h>

// PositionalEncoding add: out[e, c] = x[e, c] + PE(path_len[e], c)
//   pair index p = c - (c & 1);  div = exp(p * (-ln(10000)/512))
//   ang = path_len[e] * div;     pe  = (c even) ? sin(ang) : cos(ang)
//
// Memory-bound: 2.05 GB traffic -> ~88 us floor @ 23.3 TB/s. Strategy:
//  * 4 float4 per thread, strided by total thread count S (S % 128 == 0), so
//    col4 is constant per thread -> the two v_exp_f32 are hoisted (2 instead
//    of 8) and every b128 load/store is perfectly coalesced.
//  * row index is wave-uniform -> readfirstlane + scalar s_loads of path_len
//    (rows differ by the scalar constant S/128 between chunks).
//  * non-temporal cache policy on the streaming b128 loads/stores (read-once /
//    write-once data, keep L2 clean). Uses clang ext_vector_type, since
//    __builtin_nontemporal_* rejects HIP_vector_type classes.
//  * one clamped global_prefetch_b8 (no exec-mask branch) to warm L2.

typedef float v4f __attribute__((ext_vector_type(4)));

#define PE_THREADS 256
#define PE_ITER    4

__global__ __launch_bounds__(PE_THREADS) void pe_add4_kernel(
    const v4f* __restrict__ x4,
    const int* __restrict__ path_len,
    v4f*       __restrict__ o4,
    int nThreads,      // total threads = n4 / 4; multiple of 128
    int rowStride,     // nThreads / 128 (rows between consecutive chunks)
    int n4)            // total float4 count (for prefetch clamp)
{
    int tid = blockIdx.x * PE_THREADS + threadIdx.x;

    // col4 is identical for all 4 chunks of this thread (nThreads % 128 == 0)
    int col4 = tid & 127;
    // 32 | 128 -> row index is wave-uniform
    int row0 = __builtin_amdgcn_readfirstlane(tid) >> 7;

    // scalar loads of the four wave-uniform path lengths
    float pl0 = (float)path_len[row0];
    float pl1 = (float)path_len[row0 + rowStride];
    float pl2 = (float)path_len[row0 + 2 * rowStride];
    float pl3 = (float)path_len[row0 + 3 * rowStride];

    // hoisted div_term (hardware v_exp_f32), shared by all 4 chunks
    const float kfac = -9.21034037197618f / 512.0f;   // -ln(10000)/d_model
    float c0   = (float)(col4 << 2);
    float div0 = __expf(kfac * c0);                   // pair 4j
    float div1 = __expf(kfac * (c0 + 2.0f));          // pair 4j+2

    // clamped far prefetch (gfx1250 global_prefetch_b8), no exec branch
    int pf = tid + 3 * nThreads + (1 << 15);
    pf = pf < (n4 - 1) ? pf : (n4 - 1);
    __builtin_prefetch((const void*)(x4 + pf), 0, 1);

    const v4f* p = x4 + tid;
    v4f v0 = __builtin_nontemporal_load(p);
    v4f v1 = __builtin_nontemporal_load(p + nThreads);
    v4f v2 = __builtin_nontemporal_load(p + 2 * nThreads);
    v4f v3 = __builtin_nontemporal_load(p + 3 * nThreads);

    float a, b;
    v4f r;

    a = pl0 * div0; b = pl0 * div1;
    r.x = v0.x + __sinf(a); r.y = v0.y + __cosf(a);
    r.z = v0.z + __sinf(b); r.w = v0.w + __cosf(b);
    __builtin_nontemporal_store(r, o4 + tid);

    a = pl1 * div0; b = pl1 * div1;
    r.x = v1.x + __sinf(a); r.y = v1.y + __cosf(a);
    r.z = v1.z + __sinf(b); r.w = v1.w + __cosf(b);
    __builtin_nontemporal_store(r, o4 + tid + nThreads);

    a = pl2 * div0; b = pl2 * div1;
    r.x = v2.x + __sinf(a); r.y = v2.y + __cosf(a);
    r.z = v2.z + __sinf(b); r.w = v2.w + __cosf(b);
    __builtin_nontemporal_store(r, o4 + tid + 2 * nThreads);

    a = pl3 * div0; b = pl3 * div1;
    r.x = v3.x + __sinf(a); r.y = v3.y + __cosf(a);
    r.z = v3.z + __sinf(b); r.w = v3.w + __cosf(b);
    __builtin_nontemporal_store(r, o4 + tid + 3 * nThreads);
}

// Bounds-checked fallback (1 float4 per thread) for sizes that don't divide.
__global__ __launch_bounds__(PE_THREADS) void pe_add1_kernel(
    const v4f* __restrict__ x4,
    const int* __restrict__ path_len,
    v4f*       __restrict__ o4,
    int n4)
{
    int i = blockIdx.x * PE_THREADS + threadIdx.x;
    if (i >= n4) return;
    int row  = i >> 7;
    int col4 = i & 127;

    float pl = (float)path_len[row];
    const float kfac = -9.21034037197618f / 512.0f;
    float c0 = (float)(col4 << 2);
    float a  = pl * __expf(kfac * c0);
    float b  = pl * __expf(kfac * (c0 + 2.0f));

    v4f v = x4[i];
    v4f r;
    r.x = v.x + __sinf(a); r.y = v.y + __cosf(a);
    r.z = v.z + __sinf(b); r.w = v.w + __cosf(b);
    o4[i] = r;
}

extern "C" void kernel_launch(void* const* d_in, const int* in_sizes, int n_in,
                              void* d_out, int out_size, void* d_ws, size_t ws_size,
                              hipStream_t stream)
{
    const v4f* x4  = (const v4f*)d_in[0];  // [num_edges * 512] f32
    const int* pl  = (const int*)d_in[1];  // [num_edges] i32
    v4f*       o4  = (v4f*)d_out;          // [num_edges * 512] f32

    int n  = in_sizes[0];          // 256,000,000 elements
    int n4 = n >> 2;               // 64,000,000 float4

    if ((n4 % (PE_THREADS * PE_ITER)) == 0 && ((n4 / PE_ITER) % 128) == 0) {
        int nThreads  = n4 / PE_ITER;          // 16,000,000 (multiple of 128)
        int rowStride = nThreads >> 7;         // 125,000
        int blocks    = nThreads / PE_THREADS; // 62,500
        pe_add4_kernel<<<blocks, PE_THREADS, 0, stream>>>(
            x4, pl, o4, nThreads, rowStride, n4);
    } else {
        int blocks = (n4 + PE_THREADS - 1) / PE_THREADS;
        pe_add1_kernel<<<blocks, PE_THREADS, 0, stream>>>(x4, pl, o4, n4);
    }
}